// ArgumentScorerLearned_14439680049698
// MI455X (gfx1250) — compile-verified
//
#include <hip/hip_runtime.h>
#include <hip/hip_bf16.h>

// Problem constants from the reference
#define B_ 8
#define T_ 48
#define E_ 96
#define D_ 768    // GEMM K (compile-time for both GEMMs)
#define H_ 2048   // GEMM N (compile-time for both GEMMs)
#define O_ 2

typedef __attribute__((ext_vector_type(2))) float v2f;
typedef __attribute__((ext_vector_type(8))) float v8f;

// ---------------------------------------------------------------------------
// GEMM: C[M,2048] = A[M,768] @ W[768,2048] (+ bias), all f32 row-major.
// Uses V_WMMA_F32_16X16X4_F32 (exact f32 matrix math on CDNA5).
// Block = 128 threads = 4 waves; each wave computes one 16x16 tile,
// block covers 16 (M) x 64 (N). K=768 -> 192 WMMA steps per tile.
//
// K and N are compile-time so the inner loop is pure pointer-bump +
// immediate-offset loads: per 4-WMMA unroll window, 2 u64 pointer adds,
// 4x global_load_b64 (A), 8x global_load_b32 (W), 4x wmma. No per-load
// 64-bit address multiplies.
//
// Fragment layouts per CDNA5 ISA 7.12.2 (wave32):
//   A 16x4:  lane&15 = M, lane>>4 selects K-pair {0,1} vs {2,3}; v[0]=K, v[1]=K+1
//   B 4x16:  lane&15 = N, lane>>4 selects K-pair;                v[0]=K, v[1]=K+1
//   C/D 16x16: lane&15 = N, VGPR r -> M = r + 8*(lane>>4)
// ---------------------------------------------------------------------------
__global__ __launch_bounds__(128)
void gemm_f32_wmma(const float* __restrict__ A, const float* __restrict__ W,
                   const float* __restrict__ bias, float* __restrict__ C,
                   int has_bias) {
  const int lane = threadIdx.x & 31;
  const int wave = threadIdx.x >> 5;
  const int half = lane >> 4;   // 0/1 -> K offset 0 or 2
  const int mr   = lane & 15;   // M index for A, N index for B/D

  const int n0 = blockIdx.x * 64 + wave * 16;
  const int m0 = blockIdx.y * 16;

  const float* __restrict__ ap = A + (size_t)(m0 + mr) * D_ + 2 * half;
  const float* __restrict__ wp = W + (size_t)(2 * half) * H_ + (n0 + mr);

  v8f acc = {};
#pragma unroll 4
  for (int k0 = 0; k0 < D_; k0 += 4) {
    v2f a;
    a.x = ap[0];
    a.y = ap[1];        // contiguous -> global_load_b64, imm offset
    v2f b;
    b.x = wp[0];        // imm offset 0
    b.y = wp[H_];       // imm offset 8192 B
    // 8 args: (neg_a, A, neg_b, B, c_mod, C, reuse_a, reuse_b)
    acc = __builtin_amdgcn_wmma_f32_16x16x4_f32(
        false, a, false, b, (short)0, acc, false, false);
    ap += 4;
    wp += (size_t)4 * H_;
  }

  const float bv = has_bias ? bias[n0 + mr] : 0.0f;
  float* __restrict__ cp = C + (size_t)(m0 + 8 * half) * H_ + (n0 + mr);
#pragma unroll
  for (int r = 0; r < 8; ++r) {
    cp[(size_t)r * H_] = acc[r] + bv;   // imm offsets r*8192 B
  }
}

// ---------------------------------------------------------------------------
// Fused: out[b,t,e,o] = sum_h relu(trig_h[b,t,h] + arg_h[b,e,h]) * W2[h,o] + b2[o]
// (b1 already folded into trig_h by the first GEMM's bias epilogue.)
// Block = 256 threads = 16x16 (t,e) pairs; sweep H in 128-float chunks via LDS.
// ---------------------------------------------------------------------------
#define CH 128
#define ROWPAD 4   // LDS row stride = 132 floats (528B, 16B aligned, bank-skewed)

__global__ __launch_bounds__(256)
void fused_relu_score(const float* __restrict__ trig_h,
                      const float* __restrict__ arg_h,
                      const float* __restrict__ W2,
                      const float* __restrict__ b2,
                      float* __restrict__ out) {
  __shared__ float sh_trig[16][CH + ROWPAD];
  __shared__ float sh_arg [16][CH + ROWPAD];
  __shared__ float sh_w2  [H_ * O_];           // 16 KB

  const int b   = blockIdx.z;
  const int t0  = blockIdx.y * 16;
  const int e0  = blockIdx.x * 16;
  const int tid = threadIdx.x;
  const int t   = tid & 15;
  const int e   = tid >> 4;

  // Stage all of W2 once (4096 floats, 16 per thread).
  for (int i = tid * 4; i < H_ * O_; i += 256 * 4) {
    *(float4*)&sh_w2[i] = *(const float4*)&W2[i];
  }

  const float* __restrict__ trig_base = trig_h + (size_t)(b * T_ + t0) * H_;
  const float* __restrict__ arg_base  = arg_h  + (size_t)(b * E_ + e0) * H_;

  // Cooperative loader mapping: 256 threads move a 16x128 f32 tile as float4s.
  const int lrow = tid >> 5;        // 0..7 (and lrow+8)
  const int lcol = (tid & 31) * 4;  // 0..124

  float acc0 = 0.0f, acc1 = 0.0f;

  for (int hc = 0; hc < H_; hc += CH) {
    __syncthreads();  // protects prior chunk's readers (and W2 staging on hc==0)
    *(float4*)&sh_trig[lrow    ][lcol] = *(const float4*)&trig_base[(size_t)(lrow    ) * H_ + hc + lcol];
    *(float4*)&sh_trig[lrow + 8][lcol] = *(const float4*)&trig_base[(size_t)(lrow + 8) * H_ + hc + lcol];
    *(float4*)&sh_arg [lrow    ][lcol] = *(const float4*)&arg_base [(size_t)(lrow    ) * H_ + hc + lcol];
    *(float4*)&sh_arg [lrow + 8][lcol] = *(const float4*)&arg_base [(size_t)(lrow + 8) * H_ + hc + lcol];
    __syncthreads();

#pragma unroll 4
    for (int h = 0; h < CH; h += 4) {
      const float4 tv = *(const float4*)&sh_trig[t][h];  // broadcast across e
      const float4 av = *(const float4*)&sh_arg [e][h];  // broadcast across t
      const int hw = (hc + h) * O_;
      const float4 wA = *(const float4*)&sh_w2[hw];      // w2[h], w2[h+1]
      const float4 wB = *(const float4*)&sh_w2[hw + 4];  // w2[h+2], w2[h+3]
      float v;
      v = fmaxf(tv.x + av.x, 0.0f); acc0 = fmaf(v, wA.x, acc0); acc1 = fmaf(v, wA.y, acc1);
      v = fmaxf(tv.y + av.y, 0.0f); acc0 = fmaf(v, wA.z, acc0); acc1 = fmaf(v, wA.w, acc1);
      v = fmaxf(tv.z + av.z, 0.0f); acc0 = fmaf(v, wB.x, acc0); acc1 = fmaf(v, wB.y, acc1);
      v = fmaxf(tv.w + av.w, 0.0f); acc0 = fmaf(v, wB.z, acc0); acc1 = fmaf(v, wB.w, acc1);
    }
  }

  const size_t o = (((size_t)b * T_ + (t0 + t)) * E_ + (e0 + e)) * O_;
  out[o]     = acc0 + b2[0];
  out[o + 1] = acc1 + b2[1];
}

// ---------------------------------------------------------------------------
// Launch: GEMM trig_h = trig @ W1[:D] + b1 ; GEMM arg_h = arg @ W1[D:] ;
// fused relu/score kernel. Intermediates live in d_ws (9 MB, fits L2).
// ---------------------------------------------------------------------------
extern "C" void kernel_launch(void* const* d_in, const int* in_sizes, int n_in,
                              void* d_out, int out_size, void* d_ws, size_t ws_size,
                              hipStream_t stream) {
  const float* trig = (const float*)d_in[0];  // [B,T,D]
  const float* arg  = (const float*)d_in[1];  // [B,E,D]
  const float* W1   = (const float*)d_in[2];  // [2D,H]
  const float* b1   = (const float*)d_in[3];  // [H]
  const float* W2   = (const float*)d_in[4];  // [H,O]
  const float* b2   = (const float*)d_in[5];  // [O]
  float* out = (float*)d_out;                 // [B,T,E,O]

  float* trig_h = (float*)d_ws;                         // [B*T, H]
  float* arg_h  = trig_h + (size_t)B_ * T_ * H_;        // [B*E, H]

  dim3 blk(128);
  dim3 g1(H_ / 64, (B_ * T_) / 16);   // 32 x 24
  gemm_f32_wmma<<<g1, blk, 0, stream>>>(trig, W1, b1, trig_h, 1);
  dim3 g2(H_ / 64, (B_ * E_) / 16);   // 32 x 48
  gemm_f32_wmma<<<g2, blk, 0, stream>>>(arg, W1 + (size_t)D_ * H_, nullptr, arg_h, 0);

  dim3 g3(E_ / 16, T_ / 16, B_);      // 6 x 3 x 8
  fused_relu_score<<<g3, dim3(256), 0, stream>>>(trig_h, arg_h, W2, b2, out);
}